// HyperNetwork_FC_Fusion_VQVAE_68874095558710
// MI455X (gfx1250) — compile-verified
//
#include <hip/hip_runtime.h>
#include <hip/hip_bf16.h>
#include <math.h>

typedef __attribute__((ext_vector_type(2))) float v2f;
typedef __attribute__((ext_vector_type(8))) float v8f;

#define BQ     512
#define SQ     50
#define DQ     32
#define KQ     10000
#define NQ     (BQ * SQ)          // 25600 query rows
#define TRIG   30
#define NSPLIT 4
#define NTILES (KQ / 16)          // 625
#define TPS    ((NTILES + NSPLIT - 1) / NSPLIT)   // 157 tiles per split

// ---------------------------------------------------------------------------
// 1) |t_k|^2 per codeword
// ---------------------------------------------------------------------------
__global__ void t2_kernel(const float* __restrict__ t, float* __restrict__ t2) {
    int n = blockIdx.x * blockDim.x + threadIdx.x;
    if (n >= KQ) return;
    const float* r = t + n * DQ;
    float s = 0.f;
#pragma unroll
    for (int d = 0; d < DQ; ++d) s = fmaf(r[d], r[d], s);
    t2[n] = s;
}

// ---------------------------------------------------------------------------
// 2) Fused NN search: f32 WMMA GEMM (-2*q.t + |t|^2) + running argmin.
//    One wave handles one 16-query M-tile over one N-split of the codebook.
// ---------------------------------------------------------------------------
__global__ void nn_wmma(const float* __restrict__ q,   // (NQ, 32)
                        const float* __restrict__ t,   // (KQ, 32)
                        const float* __restrict__ t2,  // (KQ)
                        float* __restrict__ cval,      // (NQ, NSPLIT)
                        int*   __restrict__ cidx)      // (NQ, NSPLIT)
{
    const int wave  = blockIdx.x * (blockDim.x >> 5) + (threadIdx.x >> 5);
    const int lane  = threadIdx.x & 31;
    const int mtile = wave >> 2;        // / NSPLIT
    const int split = wave & (NSPLIT - 1);
    const int m0    = mtile * 16;
    const int sel   = lane >> 4;        // 0: K pair {0,1}; 1: K pair {2,3}
    const int l16   = lane & 15;

    // A fragments: 16x4 f32 tiles, K-chunks kc = 4k. Lane layout per ISA
    // (lanes 0-15: M=l16, VGPR0=K0,VGPR1=K1; lanes 16-31: K2,K3).
    const float* qrow = q + (m0 + l16) * DQ + sel * 2;
    v2f a[8];
#pragma unroll
    for (int k = 0; k < 8; ++k) a[k] = *(const v2f*)(qrow + 4 * k);

    float best[8];
    int   bidx[8];
#pragma unroll
    for (int r = 0; r < 8; ++r) { best[r] = 3.0e38f; bidx[r] = 0; }

    const int nt0 = split * TPS;
    const int nt1 = (nt0 + TPS < NTILES) ? (nt0 + TPS) : NTILES;

    // Incrementally advanced pointers: no per-iteration 64-bit address rebuild.
    const float* trow = t + (nt0 * 16 + l16) * DQ + sel * 2;
    const float* t2p  = t2 + nt0 * 16 + l16;
    int ncur = nt0 * 16 + l16;

    for (int nt = nt0; nt < nt1; ++nt) {
        // Unconditional speculative prefetch of the next tile's row: per ISA,
        // TH=0 prefetch is speculative (translation faults silently dropped),
        // so running one tile past the codebook end is safe and branch-free.
        __builtin_prefetch(trow + 16 * DQ, 0, 0);

        v2f bf[8];
#pragma unroll
        for (int k = 0; k < 8; ++k) bf[k] = *(const v2f*)(trow + 4 * k);

        float bias = *t2p;

        v8f acc = {};
#pragma unroll
        for (int k = 0; k < 8; ++k)
            acc = __builtin_amdgcn_wmma_f32_16x16x4_f32(
                false, a[k], false, bf[k], (short)0, acc, false, false);

        // acc[r] holds q.t for (M = m0 + r + 8*sel, N = ncur)
#pragma unroll
        for (int r = 0; r < 8; ++r) {
            float s = fmaf(-2.0f, acc[r], bias);   // |t|^2 - 2 q.t
            if (s < best[r]) { best[r] = s; bidx[r] = ncur; }
        }

        trow += 16 * DQ;
        t2p  += 16;
        ncur += 16;
    }

    // Reduce over the 16 N-lanes of each half-wave; first-min tie-break.
#pragma unroll
    for (int r = 0; r < 8; ++r) {
        float v  = best[r];
        int   ix = bidx[r];
#pragma unroll
        for (int m = 8; m >= 1; m >>= 1) {
            float ov = __shfl_xor(v,  m, 32);
            int   oi = __shfl_xor(ix, m, 32);
            if (ov < v || (ov == v && oi < ix)) { v = ov; ix = oi; }
        }
        if (l16 == 0) {
            int mrow = m0 + r + 8 * sel;
            cval[mrow * NSPLIT + split] = v;
            cidx[mrow * NSPLIT + split] = ix;
        }
    }
}

// ---------------------------------------------------------------------------
// 3) Combine split candidates, gather codeword, VQ loss partials, decoder.
//    256 threads = 8 rows x 32 lanes.
// ---------------------------------------------------------------------------
__global__ void recon_loss(const float* __restrict__ embd,
                           const float* __restrict__ z2,
                           const float* __restrict__ cval,
                           const int*   __restrict__ cidx,
                           const float* __restrict__ dw1, const float* __restrict__ db1,
                           const float* __restrict__ dw2, const float* __restrict__ db2,
                           float* __restrict__ zrec, float* __restrict__ lossPart)
{
    __shared__ float ze[8][32];
    __shared__ float h1[8][64];
    __shared__ float rowp[8];

    const int g    = threadIdx.x >> 5;
    const int lane = threadIdx.x & 31;
    const int row  = blockIdx.x * 8 + g;

    // argmin over split candidates (ascending split => first-min semantics)
    float bv = cval[row * NSPLIT];
    int   bi = cidx[row * NSPLIT];
#pragma unroll
    for (int s = 1; s < NSPLIT; ++s) {
        float v = cval[row * NSPLIT + s];
        int   i = cidx[row * NSPLIT + s];
        if (v < bv) { bv = v; bi = i; }
    }

    float zev = embd[bi * DQ + lane];
    ze[g][lane] = zev;

    float d = z2[row * DQ + lane] - zev;
    float p = d * d;
#pragma unroll
    for (int m = 16; m >= 1; m >>= 1) p += __shfl_xor(p, m, 32);
    if (lane == 0) rowp[g] = p;
    __syncthreads();
    if (threadIdx.x == 0) {
        float s = 0.f;
#pragma unroll
        for (int i = 0; i < 8; ++i) s += rowp[i];
        lossPart[blockIdx.x] = s;
    }

    // decoder: h1 = leaky_relu(ze @ dw1.T + db1, 0.1), 64 units
#pragma unroll
    for (int jj = 0; jj < 2; ++jj) {
        int j = lane + jj * 32;
        float acc = db1[j];
#pragma unroll
        for (int dd = 0; dd < 32; ++dd) acc = fmaf(ze[g][dd], dw1[j * 32 + dd], acc);
        h1[g][j] = acc > 0.f ? acc : 0.1f * acc;
    }
    __syncthreads();
    float acc = db2[lane];
#pragma unroll
    for (int j = 0; j < 64; ++j) acc = fmaf(h1[g][j], dw2[lane * 64 + j], acc);
    zrec[row * DQ + lane] = acc;
}

// ---------------------------------------------------------------------------
// 4) Deterministic loss reduction -> d_out[512]
// ---------------------------------------------------------------------------
__global__ void loss_reduce(const float* __restrict__ lossPart, float* __restrict__ out) {
    __shared__ float buf[256];
    float s = 0.f;
    for (int i = threadIdx.x; i < NQ / 8; i += 256) s += lossPart[i];
    buf[threadIdx.x] = s;
    __syncthreads();
    for (int m = 128; m >= 1; m >>= 1) {
        if (threadIdx.x < m) buf[threadIdx.x] += buf[threadIdx.x + m];
        __syncthreads();
    }
    if (threadIdx.x == 0) out[512] = buf[0] / (float)(BQ * SQ * DQ);
}

// ---------------------------------------------------------------------------
// 5) GRU (steps 0..30) + trigger MLP. One wave per (batch row, item|cat).
//    Blocks 0..63 = item (input zrec), 64..127 = cat (input z2).
// ---------------------------------------------------------------------------
__global__ void gru_mlp(const float* __restrict__ zrec, const float* __restrict__ z2,
                        const float* iw_ih, const float* iw_hh,
                        const float* ib_ih, const float* ib_hh,
                        const float* cw_ih, const float* cw_hh,
                        const float* cb_ih, const float* cb_hh,
                        const float* im_w1, const float* im_b1,
                        const float* im_w2, const float* im_b2,
                        const float* cm_w1, const float* cm_b1,
                        const float* cm_w2, const float* cm_b2,
                        float* __restrict__ zz1, float* __restrict__ zz2)
{
    __shared__ float Lwih[96 * 32], Lwhh[96 * 32];
    __shared__ float Lbih[96], Lbhh[96];
    __shared__ float Lm1[32 * 32], Lm2[32 * 32], Lmb1[32], Lmb2[32];

    const bool isCat = blockIdx.x >= 64;
    const float* wih = isCat ? cw_ih : iw_ih;
    const float* whh = isCat ? cw_hh : iw_hh;
    const float* bih = isCat ? cb_ih : ib_ih;
    const float* bhh = isCat ? cb_hh : ib_hh;
    const float* mw1 = isCat ? cm_w1 : im_w1;
    const float* mb1 = isCat ? cm_b1 : im_b1;
    const float* mw2 = isCat ? cm_w2 : im_w2;
    const float* mb2 = isCat ? cm_b2 : im_b2;

    for (int i = threadIdx.x; i < 96 * 32; i += blockDim.x) { Lwih[i] = wih[i]; Lwhh[i] = whh[i]; }
    for (int i = threadIdx.x; i < 96;      i += blockDim.x) { Lbih[i] = bih[i]; Lbhh[i] = bhh[i]; }
    for (int i = threadIdx.x; i < 32 * 32; i += blockDim.x) { Lm1[i] = mw1[i];  Lm2[i] = mw2[i]; }
    if (threadIdx.x < 32) { Lmb1[threadIdx.x] = mb1[threadIdx.x]; Lmb2[threadIdx.x] = mb2[threadIdx.x]; }
    __syncthreads();

    const int lane = threadIdx.x & 31;
    const int b    = (blockIdx.x & 63) * 8 + (threadIdx.x >> 5);
    const float* xsrc = isCat ? z2 : zrec;

    float h = 0.f;  // lane j holds hidden unit j
    for (int t = 0; t <= TRIG; ++t) {
        float xv  = xsrc[(b * SQ + t) * DQ + lane];
        float ir  = Lbih[lane],      iz = Lbih[lane + 32], in_ = Lbih[lane + 64];
        float hr  = Lbhh[lane],      hz = Lbhh[lane + 32], hn  = Lbhh[lane + 64];
#pragma unroll 4
        for (int i = 0; i < 32; ++i) {
            float xi = __shfl(xv, i, 32);
            float hi = __shfl(h,  i, 32);
            ir  = fmaf(xi, Lwih[lane * 32 + i],        ir);
            iz  = fmaf(xi, Lwih[(lane + 32) * 32 + i], iz);
            in_ = fmaf(xi, Lwih[(lane + 64) * 32 + i], in_);
            hr  = fmaf(hi, Lwhh[lane * 32 + i],        hr);
            hz  = fmaf(hi, Lwhh[(lane + 32) * 32 + i], hz);
            hn  = fmaf(hi, Lwhh[(lane + 64) * 32 + i], hn);
        }
        float r = 1.f / (1.f + __expf(-(ir + hr)));
        float z = 1.f / (1.f + __expf(-(iz + hz)));
        float n = tanhf(in_ + r * hn);
        h = (1.f - z) * n + z * h;
    }

    // zz = tanh(h @ w1.T + b1) @ w2.T + b2
    float a = Lmb1[lane];
#pragma unroll 4
    for (int i = 0; i < 32; ++i) a = fmaf(__shfl(h, i, 32), Lm1[lane * 32 + i], a);
    a = tanhf(a);
    float o = Lmb2[lane];
#pragma unroll 4
    for (int i = 0; i < 32; ++i) o = fmaf(__shfl(a, i, 32), Lm2[lane * 32 + i], o);
    (isCat ? zz2 : zz1)[b * DQ + lane] = o;
}

// ---------------------------------------------------------------------------
// 6) Hypernetwork FC stack. One block (64 threads) per sample.
// ---------------------------------------------------------------------------
struct HyperP  { const float *w1, *b1, *w2, *b2, *w3, *b3, *w4, *b4; };
struct HyperAll { HyperP L[3]; };

__global__ void hyper_fc(const float* __restrict__ x1,
                         const float* __restrict__ zz1,
                         const float* __restrict__ zz2,
                         HyperAll hp, float* __restrict__ out)
{
    __shared__ float s1[32], s2[32], h[64], hnext[64];
    const int b = blockIdx.x, tid = threadIdx.x;
    if (tid < 32) {
        s1[tid] = zz1[b * 32 + tid];
        s2[tid] = zz2[b * 32 + tid];
        h[tid]  = x1[b * 32 + tid];
    }
    __syncthreads();

    const int ins_[3]  = {32, 64, 32};
    const int outs_[3] = {64, 32, 1};
#pragma unroll
    for (int L = 0; L < 3; ++L) {
        const int ins = ins_[L], outs = outs_[L], io = ins * outs;
        const HyperP P = hp.L[L];
        if (tid < outs) {
            const int o = tid;
            float ib = P.b2[o], cb = P.b4[o];
            for (int d = 0; d < 32; ++d) {
                ib = fmaf(s1[d], P.w2[d * outs + o], ib);
                cb = fmaf(s2[d], P.w4[d * outs + o], cb);
            }
            float acc = fminf(fmaxf(ib, -0.01f), 0.01f) + cb;
            for (int i = 0; i < ins; ++i) {
                const int c = i * outs + o;
                float iw = P.b1[c], cw = P.b3[c];
                for (int d = 0; d < 32; ++d) {
                    iw = fmaf(s1[d], P.w1[d * io + c], iw);
                    cw = fmaf(s2[d], P.w3[d * io + c], cw);
                }
                acc = fmaf(h[i], fminf(fmaxf(iw, -0.01f), 0.01f) + cw, acc);
            }
            if (L < 2) acc = tanhf(acc);
            hnext[o] = acc;
        }
        __syncthreads();
        if (tid < outs) h[tid] = hnext[tid];
        __syncthreads();
    }
    if (tid == 0) out[b] = h[0];
}

// ---------------------------------------------------------------------------
extern "C" void kernel_launch(void* const* d_in, const int* in_sizes, int n_in,
                              void* d_out, int out_size, void* d_ws, size_t ws_size,
                              hipStream_t stream)
{
    (void)in_sizes; (void)n_in; (void)out_size; (void)ws_size;
    const float* x1   = (const float*)d_in[0];
    const float* z1   = (const float*)d_in[1];
    const float* z2   = (const float*)d_in[3];
    const float* embd = (const float*)d_in[4];
    const float* dw1  = (const float*)d_in[5];
    const float* db1  = (const float*)d_in[6];
    const float* dw2  = (const float*)d_in[7];
    const float* db2  = (const float*)d_in[8];

    // workspace layout (floats)
    float* ws    = (float*)d_ws;
    float* t2    = ws;                                  // 10240
    float* cval  = ws + 10240;                          // 102400
    int*   cidx  = (int*)(ws + 10240 + 102400);         // 102400
    float* zrec  = ws + 10240 + 204800;                 // 819200
    float* zz1w  = zrec + (size_t)NQ * DQ;              // 16384
    float* zz2w  = zz1w + BQ * DQ;                      // 16384
    float* lossP = zz2w + BQ * DQ;                      // 3200

    float* out = (float*)d_out;

    t2_kernel<<<dim3((KQ + 255) / 256), dim3(256), 0, stream>>>(embd, t2);

    // 1600 M-tiles * 4 N-splits = 6400 waves; 8 waves/block -> 800 blocks
    nn_wmma<<<dim3(800), dim3(256), 0, stream>>>(z1, embd, t2, cval, cidx);

    recon_loss<<<dim3(NQ / 8), dim3(256), 0, stream>>>(embd, z2, cval, cidx,
                                                       dw1, db1, dw2, db2, zrec, lossP);

    loss_reduce<<<dim3(1), dim3(256), 0, stream>>>(lossP, out);

    gru_mlp<<<dim3(128), dim3(256), 0, stream>>>(
        zrec, z2,
        (const float*)d_in[9],  (const float*)d_in[10], (const float*)d_in[11], (const float*)d_in[12],
        (const float*)d_in[13], (const float*)d_in[14], (const float*)d_in[15], (const float*)d_in[16],
        (const float*)d_in[17], (const float*)d_in[18], (const float*)d_in[19], (const float*)d_in[20],
        (const float*)d_in[21], (const float*)d_in[22], (const float*)d_in[23], (const float*)d_in[24],
        zz1w, zz2w);

    HyperAll hp;
    for (int L = 0; L < 3; ++L) {
        const int base = 25 + L * 8;
        hp.L[L].w1 = (const float*)d_in[base + 0];
        hp.L[L].b1 = (const float*)d_in[base + 1];
        hp.L[L].w2 = (const float*)d_in[base + 2];
        hp.L[L].b2 = (const float*)d_in[base + 3];
        hp.L[L].w3 = (const float*)d_in[base + 4];
        hp.L[L].b3 = (const float*)d_in[base + 5];
        hp.L[L].w4 = (const float*)d_in[base + 6];
        hp.L[L].b4 = (const float*)d_in[base + 7];
    }
    hyper_fc<<<dim3(BQ), dim3(64), 0, stream>>>(x1, zz1w, zz2w, hp, out);
}